// DynamicEdgeWeighter_74337293959662
// MI455X (gfx1250) — compile-verified
//
#include <hip/hip_runtime.h>

typedef float v2f __attribute__((ext_vector_type(2)));
typedef float v8f __attribute__((ext_vector_type(8)));

#define BATCH 4
#define TSTEPS 16
#define CH 16
#define DEG 16
#define LAM 0.3f
#define COS_EPS 1e-8f
#define NORM_EPS 1e-8f

// ---------------- K0: init per-batch min/max (bit-pattern uints) ----------------
__global__ void k_init_minmax(unsigned int* minmax) {
    int i = threadIdx.x;
    if (i < 2 * BATCH) {
        // even slot = min (init +inf), odd slot = max (init 0); mean_sim >= 0 so
        // uint ordering of the float bit pattern == float ordering.
        minmax[i] = (i & 1) ? 0u : 0x7F800000u;
    }
}

// ---------------- K1: feat = [mean_t, std_t], plus 1/max(||feat||, eps) ----------------
__global__ void k_feat(const float* __restrict__ x, float* __restrict__ feat,
                       float* __restrict__ invnorm, int N) {
    int gid = blockIdx.x * blockDim.x + threadIdx.x;
    if (gid >= BATCH * N) return;
    int b = gid / N;
    int n = gid - b * N;

    float s[CH], s2[CH];
#pragma unroll
    for (int c = 0; c < CH; ++c) { s[c] = 0.f; s2[c] = 0.f; }

    for (int t = 0; t < TSTEPS; ++t) {
        const float* p = x + (((size_t)(b * TSTEPS + t)) * N + n) * CH;
#pragma unroll
        for (int c = 0; c < CH; c += 4) {
            float4 v = *(const float4*)(p + c);
            s[c + 0] += v.x; s2[c + 0] += v.x * v.x;
            s[c + 1] += v.y; s2[c + 1] += v.y * v.y;
            s[c + 2] += v.z; s2[c + 2] += v.z * v.z;
            s[c + 3] += v.w; s2[c + 3] += v.w * v.w;
        }
    }

    float* f = feat + ((size_t)b * N + n) * (2 * CH);
    float norm2 = 0.f;
#pragma unroll
    for (int c = 0; c < CH; ++c) {
        float mu  = s[c] * (1.0f / TSTEPS);
        float var = fmaxf(s2[c] * (1.0f / TSTEPS) - mu * mu, 0.f);
        float sd  = sqrtf(var);
        f[c]      = mu;
        f[CH + c] = sd;
        norm2 += mu * mu + sd * sd;
    }
    invnorm[(size_t)b * N + n] = 1.0f / fmaxf(sqrtf(norm2), COS_EPS);
}

// ---------------- K2: one wave per (edge, batch); 8x v_wmma_f32_16x16x4_f32 ----------------
// D(16x16) = M(16x32) x Ct(32x16); M rows = 16 member feats, every column of Ct = center
// feat, so D[i][j] = dot(member_i, center) for all j.
__global__ void __launch_bounds__(256)
k_edge_sim(const float* __restrict__ feat, const float* __restrict__ invnorm,
           const int* __restrict__ edge_members, const int* __restrict__ edge_centers,
           float* __restrict__ mean_sim, unsigned int* __restrict__ minmax,
           int E, int N) {
    int gid    = blockIdx.x * blockDim.x + threadIdx.x;
    int wave   = gid >> 5;
    int lane   = threadIdx.x & 31;
    if (wave >= BATCH * E) return;        // wave-uniform (grid is an exact multiple anyway)
    int e = wave % E;
    int b = wave / E;

    int  lm = lane & 15;                  // member row this lane feeds into the A matrix
    bool hi = lane >= 16;                 // high half supplies K = {4s+2, 4s+3}

    int member = edge_members[e * DEG + lm];
    int center = edge_centers[e];

    const float* mf = feat + ((size_t)b * N + member) * (2 * CH);
    const float* cf = feat + ((size_t)b * N + center) * (2 * CH);
    float invm = invnorm[(size_t)b * N + member];
    float invc = invnorm[(size_t)b * N + center];

    int ko = hi ? 2 : 0;
    v8f acc = {};
#pragma unroll
    for (int s = 0; s < 8; ++s) {         // K = 32 in chunks of 4
        v2f a  = *(const v2f*)(mf + 4 * s + ko);   // A: member feats (16x4 layout)
        v2f bb = *(const v2f*)(cf + 4 * s + ko);   // B: center feat broadcast to 16 cols
        acc = __builtin_amdgcn_wmma_f32_16x16x4_f32(
            /*neg_a=*/false, a, /*neg_b=*/false, bb,
            /*c_mod=*/(short)0, acc, /*reuse_a=*/false, /*reuse_b=*/false);
    }

    // C/D layout: VGPR r, lanes 0-15 -> D[r][lane] (members 0-7); lanes 16-31 -> D[8+r].
    float ssum = 0.f;
#pragma unroll
    for (int r = 0; r < 8; ++r) {
        int   src = (hi ? 8 : 0) + r;              // lane holding that member's invnorm
        float inj = __shfl(invm, src, 32);
        float sim = acc[r] * inj * invc;
        sim = fminf(fmaxf(sim, 0.f), 1.f);
        ssum += sim;
    }
    ssum += __shfl_xor(ssum, 16, 32);              // fold members 8-15 into 0-7 half

    if (lane == 0) {
        float ms = ssum * (1.0f / DEG);            // counts = max(deg,1) = 16
        mean_sim[(size_t)b * E + e] = ms;
        unsigned int u = __float_as_uint(ms);
        atomicMin(&minmax[b * 2 + 0], u);
        atomicMax(&minmax[b * 2 + 1], u);
    }
}

// ---------------- K3: per-batch min/max normalize, scale W ----------------
__global__ void k_final(const float* __restrict__ W, const float* __restrict__ mean_sim,
                        const unsigned int* __restrict__ minmax, float* __restrict__ out,
                        int E) {
    int gid = blockIdx.x * blockDim.x + threadIdx.x;
    if (gid >= BATCH * E) return;
    int b = gid / E;
    int e = gid - b * E;
    float smin = __uint_as_float(minmax[b * 2 + 0]);
    float smax = __uint_as_float(minmax[b * 2 + 1]);
    float ms   = (mean_sim[gid] - smin) / (smax - smin + NORM_EPS);
    out[gid]   = W[e] * (1.0f + LAM * ms);
}

extern "C" void kernel_launch(void* const* d_in, const int* in_sizes, int n_in,
                              void* d_out, int out_size, void* d_ws, size_t ws_size,
                              hipStream_t stream) {
    const float* x_raw        = (const float*)d_in[0];
    // d_in[1] = H (unused by the reference result path)
    const float* W            = (const float*)d_in[2];
    const int*   edge_members = (const int*)d_in[3];
    const int*   edge_centers = (const int*)d_in[4];
    // d_in[5] = edge_offsets: arange(E+1)*16 -> deg == 16 uniformly

    const int E = in_sizes[2];                 // 50000
    const int N = in_sizes[4];                 // 50000 (edge_centers indexes nodes 0..N-1)

    // Workspace layout (floats):
    float* ws        = (float*)d_ws;
    float* feat      = ws;                                     // B*N*32
    float* invnorm   = feat + (size_t)BATCH * N * 2 * CH;      // B*N
    float* mean_sim  = invnorm + (size_t)BATCH * N;            // B*E
    unsigned int* mm = (unsigned int*)(mean_sim + (size_t)BATCH * E);  // 2*B

    k_init_minmax<<<1, 32, 0, stream>>>(mm);

    {
        int threads = BATCH * N;
        k_feat<<<(threads + 255) / 256, 256, 0, stream>>>(x_raw, feat, invnorm, N);
    }
    {
        int waves   = BATCH * E;               // one wave32 per (edge, batch)
        int blocks  = (waves + 7) / 8;         // 8 waves (256 threads) per block
        k_edge_sim<<<blocks, 256, 0, stream>>>(feat, invnorm, edge_members, edge_centers,
                                               mean_sim, mm, E, N);
    }
    {
        int threads = BATCH * E;
        k_final<<<(threads + 255) / 256, 256, 0, stream>>>(W, mean_sim, mm, (float*)d_out, E);
    }
}